// GraphSAGEBlock_66211215835633
// MI455X (gfx1250) — compile-verified
//
#include <hip/hip_runtime.h>

typedef float v2f __attribute__((ext_vector_type(2)));
typedef float v8f __attribute__((ext_vector_type(8)));

#define N_IN  64
#define N_HID 128
#define N_OUT 64

// ---------------------------------------------------------------- zero fill
__global__ void gs_zero_kernel(float* __restrict__ p, long n) {
    long i = (long)blockIdx.x * blockDim.x + threadIdx.x;
    long stride = (long)gridDim.x * blockDim.x;
    for (; i < n; i += stride) p[i] = 0.0f;
}

// ---------------------------------------------------------------- degrees
__global__ void gs_degree_kernel(const int* __restrict__ dst,
                                 float* __restrict__ cnt, int nE) {
    int e = blockIdx.x * blockDim.x + threadIdx.x;
    if (e < nE) atomicAdd(&cnt[dst[e]], 1.0f);
}

// ---------------------------------------------------------------- scatter-add
// One thread handles one edge x 4 channels (float4 gather, 4 f32 atomics).
template <int C>
__global__ void gs_scatter_kernel(const float* __restrict__ feat,
                                  const int* __restrict__ src,
                                  const int* __restrict__ dst,
                                  float* __restrict__ agg, int nE) {
    constexpr int G = C / 4;                    // float4 groups per edge
    long idx = (long)blockIdx.x * blockDim.x + threadIdx.x;
    long total = (long)nE * G;
    if (idx >= total) return;
    int e = (int)(idx / G);
    int q = (int)(idx % G);
    int s = src[e];
    int d = dst[e];
    const float4 v = *(const float4*)(feat + (long)s * C + q * 4);
    float* o = agg + (long)d * C + q * 4;
    atomicAdd(o + 0, v.x);
    atomicAdd(o + 1, v.y);
    atomicAdd(o + 2, v.z);
    atomicAdd(o + 3, v.w);
}

// ---------------------------------------------------------------- SAGE GEMM
// out[16 x NCOL] tile per block:  relu?( mean@Wl + x@Wr + b )
// One wave per 16-wide column tile; K-loop via V_WMMA_F32_16X16X4_F32.
// Requires nNodes % 16 == 0 (holds: 100000 = 16*6250).
template <int K, int NCOL, bool RELU>
__global__ void gs_sage_gemm_kernel(const float* __restrict__ x,    // [N,K]
                                    const float* __restrict__ agg,  // [N,K]
                                    const float* __restrict__ cnt,  // [N]
                                    const float* __restrict__ Wl,   // [K,NCOL]
                                    const float* __restrict__ Wr,   // [K,NCOL]
                                    const float* __restrict__ bias, // [NCOL]
                                    float* __restrict__ out,        // [N,NCOL]
                                    int nNodes) {
    constexpr int NWAVE = NCOL / 16;            // waves per block
    constexpr int LDSK  = K + 1;                // odd stride -> no bank conflicts
    __shared__ float s_mean[16 * LDSK];

    const int tid      = threadIdx.x;
    const int wave     = tid >> 5;
    const int lane     = tid & 31;
    const int nodeBase = blockIdx.x * 16;

    // Stage mean = agg / max(cnt,1) tile into LDS (cooperative).
    for (int i = tid; i < 16 * K; i += NWAVE * 32) {
        int r = i / K, c = i % K;
        int node = nodeBase + r;
        float inv = 1.0f / fmaxf(cnt[node], 1.0f);
        s_mean[r * LDSK + c] = agg[(long)node * K + c] * inv;
    }
    __syncthreads();

    const int row  = lane & 15;                 // A row / B col / D col index
    const int half = lane >> 4;                 // K-half selector
    const int col  = wave * 16 + row;           // output column

    v8f acc = {};

    // mean @ Wl
#pragma unroll
    for (int k0 = 0; k0 < K; k0 += 4) {
        const int ka = k0 + 2 * half;
        v2f a, b;
        a.x = s_mean[row * LDSK + ka];
        a.y = s_mean[row * LDSK + ka + 1];
        b.x = Wl[(long)ka * NCOL + col];
        b.y = Wl[(long)(ka + 1) * NCOL + col];
        acc = __builtin_amdgcn_wmma_f32_16x16x4_f32(false, a, false, b,
                                                    (short)0, acc, false, false);
    }

    // x @ Wr
    const float* xrow = x + (long)(nodeBase + row) * K;
#pragma unroll
    for (int k0 = 0; k0 < K; k0 += 4) {
        const int ka = k0 + 2 * half;
        v2f a, b;
        a.x = xrow[ka];
        a.y = xrow[ka + 1];
        b.x = Wr[(long)ka * NCOL + col];
        b.y = Wr[(long)(ka + 1) * NCOL + col];
        acc = __builtin_amdgcn_wmma_f32_16x16x4_f32(false, a, false, b,
                                                    (short)0, acc, false, false);
    }

    const float bv = bias[col];
#pragma unroll
    for (int r = 0; r < 8; ++r) {
        const int m = r + 8 * half;             // D row for this VGPR slot
        float v = acc[r] + bv;
        if (RELU) v = fmaxf(v, 0.0f);
        out[(long)(nodeBase + m) * NCOL + col] = v;
    }
}

// ---------------------------------------------------------------- launcher
extern "C" void kernel_launch(void* const* d_in, const int* in_sizes, int n_in,
                              void* d_out, int out_size, void* d_ws, size_t ws_size,
                              hipStream_t stream) {
    const float* x          = (const float*)d_in[0];
    const int*   edge_index = (const int*)  d_in[1];
    const float* W1l        = (const float*)d_in[2];
    const float* W1r        = (const float*)d_in[3];
    const float* b1         = (const float*)d_in[4];
    const float* W2l        = (const float*)d_in[5];
    const float* W2r        = (const float*)d_in[6];
    const float* b2         = (const float*)d_in[7];

    const int nNodes = in_sizes[0] / N_IN;      // 100000
    const int nE     = in_sizes[1] / 2;         // 1600000
    const int* src   = edge_index;              // edge_index[0,:]
    const int* dst   = edge_index + nE;         // edge_index[1,:]

    // Workspace layout (floats): agg1 | agg2 | cnt | h
    float* ws   = (float*)d_ws;
    long   off  = 0;
    float* agg1 = ws + off; off += (long)nNodes * N_IN;    // N*64
    float* agg2 = ws + off; off += (long)nNodes * N_HID;   // N*128
    float* cnt  = ws + off; off += nNodes;                 // N
    float* h    = ws + off;                                // N*128 (fully overwritten)

    // 1) zero agg1 + agg2 + cnt (contiguous prefix)
    const long nz = (long)nNodes * (N_IN + N_HID) + nNodes;
    gs_zero_kernel<<<2048, 256, 0, stream>>>(ws, nz);

    // 2) degrees (shared by both layers)
    gs_degree_kernel<<<(nE + 255) / 256, 256, 0, stream>>>(dst, cnt, nE);

    // 3) layer-1 scatter: agg1 += x[src]
    {
        const long work = (long)nE * (N_IN / 4);
        gs_scatter_kernel<N_IN><<<(int)((work + 255) / 256), 256, 0, stream>>>(
            x, src, dst, agg1, nE);
    }

    // 4) layer-1 GEMM: h = relu(mean1@W1l + x@W1r + b1)   [8 waves/block]
    gs_sage_gemm_kernel<N_IN, N_HID, true>
        <<<nNodes / 16, (N_HID / 16) * 32, 0, stream>>>(
            x, agg1, cnt, W1l, W1r, b1, h, nNodes);

    // 5) layer-2 scatter: agg2 += h[src]
    {
        const long work = (long)nE * (N_HID / 4);
        gs_scatter_kernel<N_HID><<<(int)((work + 255) / 256), 256, 0, stream>>>(
            h, src, dst, agg2, nE);
    }

    // 6) layer-2 GEMM: out = mean2@W2l + h@W2r + b2        [4 waves/block]
    gs_sage_gemm_kernel<N_HID, N_OUT, false>
        <<<nNodes / 16, (N_OUT / 16) * 32, 0, stream>>>(
            h, agg2, cnt, W2l, W2r, b2, (float*)d_out, nNodes);
}